// GraphSage_83769042141372
// MI455X (gfx1250) — compile-verified
//
#include <hip/hip_runtime.h>
#include <hip/hip_bf16.h>
#include <math.h>

#define N_NODES 50000
#define N_EDGES 800000
#define F_IN    128
#define HID     256
#define N_CLASS 40

typedef __attribute__((ext_vector_type(2))) float v2f;
typedef __attribute__((ext_vector_type(8))) float v8f;

// ---------------------------------------------------------------------------
// 0) zero the per-node degree counters (only buffer needing init)
// ---------------------------------------------------------------------------
__global__ __launch_bounds__(256) void zero_i32(int* __restrict__ p, int n) {
  int i = blockIdx.x * blockDim.x + threadIdx.x;
  if (i < n) p[i] = 0;
}

// ---------------------------------------------------------------------------
// 1) CSR build pass 1: per-destination degree count (1.6M cheap int atomics
//    instead of 307M f32 atomics in the naive scatter formulation)
// ---------------------------------------------------------------------------
__global__ __launch_bounds__(256) void count_deg(const int* __restrict__ ei,
                                                 int* __restrict__ cnt) {
  int e = blockIdx.x * blockDim.x + threadIdx.x;
  if (e < N_EDGES) atomicAdd(&cnt[ei[N_EDGES + e]], 1);
}

// ---------------------------------------------------------------------------
// 2) CSR build pass 2: exclusive prefix scan of 50000 counts.
//    Single 1024-thread block; wave32 shuffle scan + cross-wave scan in LDS.
// ---------------------------------------------------------------------------
__global__ __launch_bounds__(1024) void scan_offsets(const int* __restrict__ cnt,
                                                     int* __restrict__ off,
                                                     int* __restrict__ cursor) {
  __shared__ int wsums[32];
  __shared__ int carry_s;
  int lane = threadIdx.x & 31;
  int wid  = threadIdx.x >> 5;
  if (threadIdx.x == 0) carry_s = 0;
  __syncthreads();

  for (int base = 0; base < N_NODES; base += 1024) {
    int i = base + threadIdx.x;
    int v = (i < N_NODES) ? cnt[i] : 0;

    // wave-inclusive scan (wave32)
    int s = v;
#pragma unroll
    for (int d = 1; d < 32; d <<= 1) {
      int t = __shfl_up(s, d, 32);
      if (lane >= d) s += t;
    }
    if (lane == 31) wsums[wid] = s;
    __syncthreads();
    if (threadIdx.x < 32) {           // wave 0 scans the 32 wave totals
      int wv = wsums[threadIdx.x];
      int ws = wv;
#pragma unroll
      for (int d = 1; d < 32; d <<= 1) {
        int t = __shfl_up(ws, d, 32);
        if (threadIdx.x >= (unsigned)d) ws += t;
      }
      wsums[threadIdx.x] = ws - wv;   // exclusive wave offset
    }
    __syncthreads();

    int incl  = s + wsums[wid];       // inclusive within chunk
    int carry = carry_s;
    int exclg = carry + incl - v;     // global exclusive
    if (i < N_NODES) { off[i] = exclg; cursor[i] = exclg; }
    __syncthreads();                  // everyone read carry_s
    if (threadIdx.x == 1023) carry_s = carry + incl;
    __syncthreads();
  }
  if (threadIdx.x == 0) off[N_NODES] = carry_s;   // == N_EDGES
}

// ---------------------------------------------------------------------------
// 3) CSR build pass 3: bucket-fill source lists via per-dst cursors
// ---------------------------------------------------------------------------
__global__ __launch_bounds__(256) void fill_src(const int* __restrict__ ei,
                                                int* __restrict__ cursor,
                                                int* __restrict__ srcl) {
  int e = blockIdx.x * blockDim.x + threadIdx.x;
  if (e < N_EDGES) {
    int dst = ei[N_EDGES + e];
    int pos = atomicAdd(&cursor[dst], 1);
    srcl[pos] = ei[e];
  }
}

// ---------------------------------------------------------------------------
// 4) gather-mean, layer 1: one wave per node, lane = feature (coalesced
//    128B reads/step, all hits in 192MB L2, zero atomics, fused 1/deg)
// ---------------------------------------------------------------------------
__global__ __launch_bounds__(256) void gather_mean_l1(const int* __restrict__ off,
                                                      const int* __restrict__ srcl,
                                                      const float* __restrict__ x,
                                                      float* __restrict__ mean1) {
  int node = (blockIdx.x * blockDim.x + threadIdx.x) >> 5;
  int lane = threadIdx.x & 31;
  if (node >= N_NODES) return;
  int e0 = off[node], e1 = off[node + 1];
  float s0 = 0.f, s1 = 0.f, s2 = 0.f, s3 = 0.f;
  for (int e = e0; e < e1; ++e) {
    const float* xs = x + (long long)srcl[e] * F_IN;
    s0 += xs[lane];
    s1 += xs[lane + 32];
    s2 += xs[lane + 64];
    s3 += xs[lane + 96];
  }
  float rd = 1.0f / fmaxf((float)(e1 - e0), 1.0f);
  float* mp = mean1 + (long long)node * F_IN;
  mp[lane]      = s0 * rd;
  mp[lane + 32] = s1 * rd;
  mp[lane + 64] = s2 * rd;
  mp[lane + 96] = s3 * rd;
}

// ---------------------------------------------------------------------------
// 5) layer-1 GEMM via V_WMMA_F32_16X16X4_F32 (pure load+wmma inner loop):
//    h = relu( mean1 @ W1l^T + b1 + x @ W1r^T )
// ---------------------------------------------------------------------------
__global__ __launch_bounds__(256) void sage_layer1(const float* __restrict__ mean1,
                                                   const float* __restrict__ x,
                                                   const float* __restrict__ W1l,
                                                   const float* __restrict__ b1,
                                                   const float* __restrict__ W1r,
                                                   float* __restrict__ h) {
  int wave  = (blockIdx.x * blockDim.x + threadIdx.x) >> 5;
  int lane  = threadIdx.x & 31;
  int tileM = wave >> 4;              // 0..3124
  int tileN = wave & 15;              // 0..15
  int half  = lane >> 4;              // K sub-pair select (f32 fragment layout)
  int m = tileM * 16 + (lane & 15);
  int n = tileN * 16 + (lane & 15);

  const float* am = mean1 + (long long)m * F_IN;
  const float* xm = x     + (long long)m * F_IN;
  const float* wl = W1l   + (long long)n * F_IN;   // B[k][n] = W1l[n][k]
  const float* wr = W1r   + (long long)n * F_IN;

  v8f acc = {};
#pragma unroll 4
  for (int k0 = 0; k0 < F_IN; k0 += 4) {
    int ka = k0 + half * 2;                        // even -> 8B aligned float2
    v2f a0 = *(const v2f*)(am + ka);
    v2f b0 = *(const v2f*)(wl + ka);
    acc = __builtin_amdgcn_wmma_f32_16x16x4_f32(false, a0, false, b0,
                                                (short)0, acc, false, false);
    v2f a1 = *(const v2f*)(xm + ka);
    v2f b1v = *(const v2f*)(wr + ka);
    acc = __builtin_amdgcn_wmma_f32_16x16x4_f32(false, a1, false, b1v,
                                                (short)0, acc, false, false);
  }

  float bias = b1[n];
#pragma unroll
  for (int r = 0; r < 8; ++r) {                    // C/D: row = r + 8*half
    int row = tileM * 16 + r + 8 * half;
    h[(long long)row * HID + n] = fmaxf(acc[r] + bias, 0.0f);
  }
}

// ---------------------------------------------------------------------------
// 6) gather-mean, layer 2 (256 features, 8 accumulators per lane)
// ---------------------------------------------------------------------------
__global__ __launch_bounds__(256) void gather_mean_l2(const int* __restrict__ off,
                                                      const int* __restrict__ srcl,
                                                      const float* __restrict__ h,
                                                      float* __restrict__ mean2) {
  int node = (blockIdx.x * blockDim.x + threadIdx.x) >> 5;
  int lane = threadIdx.x & 31;
  if (node >= N_NODES) return;
  int e0 = off[node], e1 = off[node + 1];
  float s[8] = {0.f, 0.f, 0.f, 0.f, 0.f, 0.f, 0.f, 0.f};
  for (int e = e0; e < e1; ++e) {
    const float* hs = h + (long long)srcl[e] * HID;
#pragma unroll
    for (int j = 0; j < 8; ++j) s[j] += hs[lane + 32 * j];
  }
  float rd = 1.0f / fmaxf((float)(e1 - e0), 1.0f);
  float* mp = mean2 + (long long)node * HID;
#pragma unroll
  for (int j = 0; j < 8; ++j) mp[lane + 32 * j] = s[j] * rd;
}

// ---------------------------------------------------------------------------
// 7) layer-2 GEMM via WMMA: logits = mean2 @ W2l^T + b2 + h @ W2r^T
//    40 classes -> 3 N-tiles (clamped loads, masked stores; WMMA stays in
//    uniform control flow so EXEC is all ones)
// ---------------------------------------------------------------------------
__global__ __launch_bounds__(256) void sage_layer2(const float* __restrict__ mean2,
                                                   const float* __restrict__ h,
                                                   const float* __restrict__ W2l,
                                                   const float* __restrict__ b2,
                                                   const float* __restrict__ W2r,
                                                   float* __restrict__ logits) {
  const int NT = 3;                                // ceil(40/16)
  int wave = (blockIdx.x * blockDim.x + threadIdx.x) >> 5;
  if (wave >= 3125 * NT) return;
  int lane  = threadIdx.x & 31;
  int tileM = wave / NT;
  int tileN = wave % NT;
  int half  = lane >> 4;
  int m  = tileM * 16 + (lane & 15);
  int n  = tileN * 16 + (lane & 15);
  int ns = n < N_CLASS ? n : (N_CLASS - 1);        // clamp, no divergence

  const float* am = mean2 + (long long)m * HID;
  const float* hm = h     + (long long)m * HID;
  const float* wl = W2l   + (long long)ns * HID;
  const float* wr = W2r   + (long long)ns * HID;

  v8f acc = {};
#pragma unroll 4
  for (int k0 = 0; k0 < HID; k0 += 4) {
    int ka = k0 + half * 2;
    v2f a0 = *(const v2f*)(am + ka);
    v2f b0 = *(const v2f*)(wl + ka);
    acc = __builtin_amdgcn_wmma_f32_16x16x4_f32(false, a0, false, b0,
                                                (short)0, acc, false, false);
    v2f a1 = *(const v2f*)(hm + ka);
    v2f b1v = *(const v2f*)(wr + ka);
    acc = __builtin_amdgcn_wmma_f32_16x16x4_f32(false, a1, false, b1v,
                                                (short)0, acc, false, false);
  }

  float bias = b2[ns];
  if (n < N_CLASS) {
#pragma unroll
    for (int r = 0; r < 8; ++r) {
      int row = tileM * 16 + r + 8 * half;
      logits[(long long)row * N_CLASS + n] = acc[r] + bias;
    }
  }
}

// ---------------------------------------------------------------------------
// 8) log-softmax over 40 classes; one wave32 per node, shuffle reductions
// ---------------------------------------------------------------------------
__global__ __launch_bounds__(256) void logsoftmax40(const float* __restrict__ logits,
                                                    float* __restrict__ out) {
  int wave = (blockIdx.x * blockDim.x + threadIdx.x) >> 5;
  int lane = threadIdx.x & 31;
  if (wave >= N_NODES) return;
  const float* row = logits + (long long)wave * N_CLASS;
  float v0 = row[lane];                                     // lane < 32 < 40
  float v1 = (32 + lane) < N_CLASS ? row[32 + lane] : -__builtin_inff();
  float mx = fmaxf(v0, v1);
#pragma unroll
  for (int s = 16; s; s >>= 1) mx = fmaxf(mx, __shfl_xor(mx, s, 32));
  float e = __expf(v0 - mx) + ((32 + lane) < N_CLASS ? __expf(v1 - mx) : 0.0f);
#pragma unroll
  for (int s = 16; s; s >>= 1) e += __shfl_xor(e, s, 32);
  float lse = mx + __logf(e);
  float* orow = out + (long long)wave * N_CLASS;
  orow[lane] = v0 - lse;
  if (32 + lane < N_CLASS) orow[32 + lane] = v1 - lse;
}

// ---------------------------------------------------------------------------
// launch
// ---------------------------------------------------------------------------
extern "C" void kernel_launch(void* const* d_in, const int* in_sizes, int n_in,
                              void* d_out, int out_size, void* d_ws, size_t ws_size,
                              hipStream_t stream) {
  const float* x   = (const float*)d_in[0];
  const int*   ei  = (const int*)d_in[1];   // edge_index (harness integer conv)
  const float* W1l = (const float*)d_in[2];
  const float* b1  = (const float*)d_in[3];
  const float* W1r = (const float*)d_in[4];
  const float* W2l = (const float*)d_in[5];
  const float* b2  = (const float*)d_in[6];
  const float* W2r = (const float*)d_in[7];
  float* out = (float*)d_out;

  // ws layout: floats first (8B-aligned float2 loads), then CSR ints.
  float* ws     = (float*)d_ws;
  float* mean1  = ws;                                     // 50000*128
  float* hbuf   = mean1  + (size_t)N_NODES * F_IN;        // 50000*256
  float* mean2  = hbuf   + (size_t)N_NODES * HID;         // 50000*256
  float* logits = mean2  + (size_t)N_NODES * HID;         // 50000*40
  int*   cnt    = (int*)(logits + (size_t)N_NODES * N_CLASS);  // 50000
  int*   off    = cnt    + N_NODES;                       // 50001
  int*   cursor = off    + N_NODES + 1;                   // 50000
  int*   srcl   = cursor + N_NODES;                       // 800000

  // CSR build (per call; every buffer fully rewritten -> no stale state)
  zero_i32<<<(N_NODES + 255) / 256, 256, 0, stream>>>(cnt, N_NODES);
  count_deg<<<N_EDGES / 256, 256, 0, stream>>>(ei, cnt);
  scan_offsets<<<1, 1024, 0, stream>>>(cnt, off, cursor);
  fill_src<<<N_EDGES / 256, 256, 0, stream>>>(ei, cursor, srcl);

  // layer 1: gather-mean then WMMA GEMM (+bias, ReLU)
  gather_mean_l1<<<N_NODES / 8, 256, 0, stream>>>(off, srcl, x, mean1);
  sage_layer1<<<(N_NODES * 16) / 8 / 16 * 16 / 16, 256, 0, stream>>>(  // 6250
      mean1, x, W1l, b1, W1r, hbuf);

  // layer 2: gather-mean on h then WMMA GEMM into 40 classes
  gather_mean_l2<<<N_NODES / 8, 256, 0, stream>>>(off, srcl, hbuf, mean2);
  sage_layer2<<<(3125 * 3 + 7) / 8, 256, 0, stream>>>(mean2, hbuf, W2l, b2, W2r, logits);

  // final log-softmax
  logsoftmax40<<<N_NODES / 8, 256, 0, stream>>>(logits, out);
}